// GINMolHIV_43843026157808
// MI455X (gfx1250) — compile-verified
//
#include <hip/hip_runtime.h>
#include <hip/hip_bf16.h>

typedef __attribute__((ext_vector_type(2))) float v2f;
typedef __attribute__((ext_vector_type(8))) float v8f;

#define NN 100000
#define NE 200000
#define NG 5000
#define DD 300
#define D2 600
#define HLD 320   // padded stride for [N, 300] activation buffers (20 * 16)
#define TLD 640   // padded stride for [N, 600] hidden buffer (40 * 16)

// ---------------- Atom encoder: h[n][d] = sum_f atom_tables[f][x[n][f]][d]
__global__ void k_atom(const int* __restrict__ x, const float* __restrict__ at,
                       float* __restrict__ h) {
  int n = blockIdx.x;
  int d = threadIdx.x;
  if (d >= DD) return;
  float acc = 0.f;
#pragma unroll
  for (int f = 0; f < 9; ++f) {
    int id = x[n * 9 + f];
    acc += at[(size_t)(f * 128 + id) * DD + d];
  }
  h[(size_t)n * HLD + d] = acc;
}

// ---------------- z = (1 + eps[layer]) * h   (valid columns only)
__global__ void k_zinit(const float* __restrict__ h, float* __restrict__ z,
                        const float* __restrict__ eps, int li, int n) {
  int i = blockIdx.x * blockDim.x + threadIdx.x;
  if (i >= n) return;
  int row = i / DD;
  int col = i - row * DD;
  size_t a = (size_t)row * HLD + col;
  float s = 1.0f + eps[li];
  z[a] = s * h[a];
}

// ---------------- edge message + scatter-add: z[dst] += relu(h[src] + e)
__global__ void k_edge(const int* __restrict__ ei, const int* __restrict__ ea,
                       const float* __restrict__ bt, const float* __restrict__ h,
                       float* __restrict__ z) {
  int e = blockIdx.x;
  int d = threadIdx.x;
  if (d >= DD) return;
  int src = ei[e];
  int dst = ei[NE + e];
  int a0 = ea[e * 3 + 0];
  int a1 = ea[e * 3 + 1];
  int a2 = ea[e * 3 + 2];
  float m = h[(size_t)src * HLD + d]
          + bt[(size_t)(a0)      * DD + d]
          + bt[(size_t)(16 + a1) * DD + d]
          + bt[(size_t)(32 + a2) * DD + d];
  m = fmaxf(m, 0.f);
  atomicAdd(&z[(size_t)dst * HLD + d], m);
}

// ---------------- pack W[K,N] -> Bp[K/2][Npad][2] (even/odd K pair interleaved,
// zero-padded columns) so one b64 load feeds a WMMA B operand pair.
__global__ void k_packB(const float* __restrict__ W, float* __restrict__ Bp,
                        int K, int N, int Npad) {
  int idx = blockIdx.x * blockDim.x + threadIdx.x;
  int total = (K >> 1) * Npad;
  if (idx >= total) return;
  int kp = idx / Npad;
  int n  = idx - kp * Npad;
  float v0 = (n < N) ? W[(size_t)(2 * kp)     * N + n] : 0.f;
  float v1 = (n < N) ? W[(size_t)(2 * kp + 1) * N + n] : 0.f;
  size_t o = ((size_t)kp * Npad + n) * 2;
  Bp[o + 0] = v0;
  Bp[o + 1] = v1;
}

__global__ void k_packbias(const float* __restrict__ b, float* __restrict__ bp,
                           int N, int Npad) {
  int i = blockIdx.x * blockDim.x + threadIdx.x;
  if (i < Npad) bp[i] = (i < N) ? b[i] : 0.f;
}

// ---------------- fp32 WMMA GEMM: C = op(A) @ B + bias, no bounds checks.
// op(A) = A, or relu(A*scale[k]+shift[k]) (fused BN+ReLU of hidden layer).
// One wave owns a 16x64 tile: 4 x v8f accumulators, grid-stride over tiles.
template <bool BN_A>
__global__ void k_gemm(const float* __restrict__ A, int lda,
                       const float* __restrict__ scale, const float* __restrict__ shift,
                       const float* __restrict__ Bp, int Npad,
                       const float* __restrict__ biasp,
                       float* __restrict__ C, int ldc,
                       int tilesM, int tilesN64, int K) {
  int lane = threadIdx.x & 31;
  int wave = blockIdx.x * (blockDim.x >> 5) + (threadIdx.x >> 5);
  int nwaves = gridDim.x * (blockDim.x >> 5);
  int m16 = lane & 15;            // A row in tile / B,C column in sub-tile
  int kb2 = (lane >> 4) << 1;     // K phase: lanes 0-15 -> k0,k0+1; 16-31 -> k0+2,k0+3
  int total = tilesM * tilesN64;

  for (int t = wave; t < total; t += nwaves) {
    int tm = t / tilesN64;
    int tn = t - tm * tilesN64;
    int rowBase = tm * 16;
    int colBase = tn * 64;
    const float* Arow = A + (size_t)(rowBase + m16) * lda;

    v8f acc0 = v8f{}, acc1 = v8f{}, acc2 = v8f{}, acc3 = v8f{};

    for (int k0 = 0; k0 < K; k0 += 4) {
      int kb = k0 + kb2;                 // always even
      v2f a = *(const v2f*)(Arow + kb);
      if (BN_A) {
        v2f sc = *(const v2f*)(scale + kb);
        v2f sh = *(const v2f*)(shift + kb);
        a.x = fmaxf(a.x * sc.x + sh.x, 0.f);
        a.y = fmaxf(a.y * sc.y + sh.y, 0.f);
      }
      const float* Brow = Bp + ((size_t)(kb >> 1) * Npad + colBase + m16) * 2;
      v2f b0 = *(const v2f*)(Brow + 0);
      v2f b1 = *(const v2f*)(Brow + 32);
      v2f b2 = *(const v2f*)(Brow + 64);
      v2f b3 = *(const v2f*)(Brow + 96);
      acc0 = __builtin_amdgcn_wmma_f32_16x16x4_f32(false, a, false, b0, (short)0, acc0, false, false);
      acc1 = __builtin_amdgcn_wmma_f32_16x16x4_f32(false, a, false, b1, (short)0, acc1, false, false);
      acc2 = __builtin_amdgcn_wmma_f32_16x16x4_f32(false, a, false, b2, (short)0, acc2, false, false);
      acc3 = __builtin_amdgcn_wmma_f32_16x16x4_f32(false, a, false, b3, (short)0, acc3, false, false);
    }

    // epilogue: bias + store. C/D layout: vgpr r -> row r + 8*(lane>>4), col = lane&15.
    int rofs = (lane >> 4) * 8;
    int nc = colBase + m16;
    float bv0 = biasp[nc];
    float bv1 = biasp[nc + 16];
    float bv2 = biasp[nc + 32];
    float bv3 = biasp[nc + 48];
    float* Cr = C + (size_t)(rowBase + rofs) * ldc + nc;
#pragma unroll
    for (int r = 0; r < 8; ++r) {
      Cr[0]  = acc0[r] + bv0;
      Cr[16] = acc1[r] + bv1;
      Cr[32] = acc2[r] + bv2;
      Cr[48] = acc3[r] + bv3;
      Cr += ldc;
    }
  }
}

// ---------------- column sum / sumsq (training-mode BatchNorm stats)
__global__ void k_colstat(const float* __restrict__ X, int ld, int M, int N,
                          float* __restrict__ sums, float* __restrict__ sumsq) {
  int c = blockIdx.x * blockDim.x + threadIdx.x;
  if (c >= N) return;
  int chunk = (M + gridDim.y - 1) / gridDim.y;
  int r0 = blockIdx.y * chunk;
  int r1 = min(M, r0 + chunk);
  float s = 0.f, q = 0.f;
  for (int r = r0; r < r1; ++r) {
    float v = X[(size_t)r * ld + c];
    s += v;
    q += v * v;
  }
  atomicAdd(&sums[c], s);
  atomicAdd(&sumsq[c], q);
}

// ---------------- stats -> affine scale/shift: y = x*scale + shift
__global__ void k_finalize(const float* __restrict__ sums, const float* __restrict__ sumsq,
                           const float* __restrict__ g, const float* __restrict__ be,
                           float* __restrict__ scale, float* __restrict__ shift,
                           int N, float invM) {
  int c = blockIdx.x * blockDim.x + threadIdx.x;
  if (c >= N) return;
  float mean = sums[c] * invM;
  float var  = sumsq[c] * invM - mean * mean;
  float sc = g[c] * rsqrtf(var + 1e-5f);
  scale[c] = sc;
  shift[c] = be[c] - mean * sc;
}

// ---------------- apply BN (+optional ReLU) in place on valid columns
__global__ void k_apply(float* __restrict__ h, const float* __restrict__ scale,
                        const float* __restrict__ shift, int n, int relu) {
  int i = blockIdx.x * blockDim.x + threadIdx.x;
  if (i >= n) return;
  int row = i / DD;
  int col = i - row * DD;
  size_t a = (size_t)row * HLD + col;
  float v = h[a] * scale[col] + shift[col];
  if (relu) v = fmaxf(v, 0.f);
  h[a] = v;
}

// ---------------- segment-sum pooling (atomics); counts as float
__global__ void k_pool(const float* __restrict__ h, const int* __restrict__ batch,
                       float* __restrict__ sums, float* __restrict__ cnt) {
  int n = blockIdx.x;
  int d = threadIdx.x;
  int g = batch[n];
  if (d < DD) atomicAdd(&sums[(size_t)g * DD + d], h[(size_t)n * HLD + d]);
  if (d == 0) atomicAdd(&cnt[g], 1.f);
}

// ---------------- head: out[g] = relu(mean(h_g) @ Wp1 + bp1) @ Wp2 + bp2
__global__ void k_head(const float* __restrict__ sums, const float* __restrict__ cnt,
                       const float* __restrict__ Wp1, const float* __restrict__ bp1,
                       const float* __restrict__ Wp2, const float* __restrict__ bp2,
                       float* __restrict__ out) {
  __shared__ float hg[DD];
  __shared__ float part[256];
  int g = blockIdx.x;
  float ic = 1.0f / fmaxf(cnt[g], 1.0f);
  for (int d = threadIdx.x; d < DD; d += blockDim.x)
    hg[d] = sums[(size_t)g * DD + d] * ic;
  __syncthreads();
  int j = threadIdx.x;
  float p = 0.f;
  if (j < 150) {
    float acc = bp1[j];
    for (int k = 0; k < DD; ++k) acc += hg[k] * Wp1[k * 150 + j];
    p = fmaxf(acc, 0.f) * Wp2[j];
  }
  part[j] = p;
  __syncthreads();
  for (int s = 128; s > 0; s >>= 1) {
    if (j < s) part[j] += part[j + s];
    __syncthreads();
  }
  if (j == 0) out[g] = part[0] + bp2[0];
}

static inline int cdiv(int a, int b) { return (a + b - 1) / b; }

extern "C" void kernel_launch(void* const* d_in, const int* in_sizes, int n_in,
                              void* d_out, int out_size, void* d_ws, size_t ws_size,
                              hipStream_t stream) {
  (void)in_sizes; (void)n_in; (void)out_size; (void)ws_size;
  const int*   x           = (const int*)d_in[0];
  const int*   edge_index  = (const int*)d_in[1];
  const int*   edge_attr   = (const int*)d_in[2];
  const int*   batch       = (const int*)d_in[3];
  const float* atom_tables = (const float*)d_in[4];
  const float* bond_tables = (const float*)d_in[5];
  const float* eps         = (const float*)d_in[6];
  const float* W1  = (const float*)d_in[7];
  const float* b1  = (const float*)d_in[8];
  const float* g1  = (const float*)d_in[9];
  const float* be1 = (const float*)d_in[10];
  const float* W2  = (const float*)d_in[11];
  const float* b2  = (const float*)d_in[12];
  const float* g2  = (const float*)d_in[13];
  const float* be2 = (const float*)d_in[14];
  const float* Wp1 = (const float*)d_in[15];
  const float* bp1 = (const float*)d_in[16];
  const float* Wp2 = (const float*)d_in[17];
  const float* bp2 = (const float*)d_in[18];
  float* out = (float*)d_out;

  // workspace layout (floats)
  float* ws   = (float*)d_ws;
  float* h    = ws;                          // NN*HLD
  float* z    = h   + (size_t)NN * HLD;      // NN*HLD
  float* t    = z   + (size_t)NN * HLD;      // NN*TLD
  float* W1p  = t   + (size_t)NN * TLD;      // 300*640 packed
  float* W2p  = W1p + (size_t)DD * TLD;      // 600*320 packed
  float* b1p  = W2p + (size_t)D2 * HLD;      // 640
  float* b2p  = b1p + TLD;                   // 320
  float* s1   = b2p + HLD;                   // 640
  float* q1   = s1 + 640;                    // contiguous with s1
  float* sc1  = q1 + 640;
  float* sh1  = sc1 + 640;
  float* s2   = sh1 + 640;
  float* q2   = s2 + 640;                    // contiguous with s2
  float* sc2  = q2 + 640;
  float* sh2  = sc2 + 640;
  float* psum = sh2 + 640;                   // NG*DD
  float* pcnt = psum + (size_t)NG * DD;      // NG (contiguous with psum)

  const int TPB = 256;
  const int GEMM_BLOCKS = 2048;              // 8 waves/block, grid-stride over tiles
  const int tilesM = NN / 16;                // 6250, exact

  // 1) atom encoder
  k_atom<<<NN, 320, 0, stream>>>(x, atom_tables, h);

  // 2) layers
  for (int i = 0; i < 5; ++i) {
    // z = (1+eps)*h, then scatter-add edge messages
    k_zinit<<<cdiv(NN * DD, TPB), TPB, 0, stream>>>(h, z, eps, i, NN * DD);
    k_edge<<<NE, 320, 0, stream>>>(edge_index, edge_attr,
                                   bond_tables + (size_t)i * 3 * 16 * DD, h, z);

    // pack this layer's weights/biases (padded, K-pair interleaved)
    k_packB<<<cdiv((DD / 2) * TLD, TPB), TPB, 0, stream>>>(
        W1 + (size_t)i * DD * D2, W1p, DD, D2, TLD);
    k_packB<<<cdiv((D2 / 2) * HLD, TPB), TPB, 0, stream>>>(
        W2 + (size_t)i * D2 * DD, W2p, D2, DD, HLD);
    k_packbias<<<cdiv(TLD, TPB), TPB, 0, stream>>>(b1 + (size_t)i * D2, b1p, D2, TLD);
    k_packbias<<<cdiv(HLD, TPB), TPB, 0, stream>>>(b2 + (size_t)i * DD, b2p, DD, HLD);

    // GEMM1: t[NN x 640] = z @ W1_i + b1_i
    hipMemsetAsync(s1, 0, 2 * 640 * sizeof(float), stream);
    k_gemm<false><<<GEMM_BLOCKS, TPB, 0, stream>>>(
        z, HLD, nullptr, nullptr, W1p, TLD, b1p, t, TLD, tilesM, TLD / 64, DD);

    // BN1 stats -> scale/shift
    dim3 sgrid1(cdiv(D2, TPB), 128);
    k_colstat<<<sgrid1, TPB, 0, stream>>>(t, TLD, NN, D2, s1, q1);
    k_finalize<<<cdiv(D2, TPB), TPB, 0, stream>>>(s1, q1, g1 + (size_t)i * D2,
                                                  be1 + (size_t)i * D2, sc1, sh1,
                                                  D2, 1.0f / NN);

    // GEMM2: h[NN x 320] = relu(bn(t)) @ W2_i + b2_i
    hipMemsetAsync(s2, 0, 2 * 640 * sizeof(float), stream);
    k_gemm<true><<<GEMM_BLOCKS, TPB, 0, stream>>>(
        t, TLD, sc1, sh1, W2p, HLD, b2p, h, HLD, tilesM, HLD / 64, D2);

    // BN2 stats -> apply in place (+relu except last layer)
    dim3 sgrid2(cdiv(DD, TPB), 128);
    k_colstat<<<sgrid2, TPB, 0, stream>>>(h, HLD, NN, DD, s2, q2);
    k_finalize<<<cdiv(DD, TPB), TPB, 0, stream>>>(s2, q2, g2 + (size_t)i * DD,
                                                  be2 + (size_t)i * DD, sc2, sh2,
                                                  DD, 1.0f / NN);
    k_apply<<<cdiv(NN * DD, TPB), TPB, 0, stream>>>(h, sc2, sh2, NN * DD,
                                                    (i < 4) ? 1 : 0);
  }

  // 3) pooling + head
  hipMemsetAsync(psum, 0, ((size_t)NG * DD + NG) * sizeof(float), stream);
  k_pool<<<NN, 320, 0, stream>>>(h, batch, psum, pcnt);
  k_head<<<NG, 256, 0, stream>>>(psum, pcnt, Wp1, bp1, Wp2, bp2, out);
}